// EmbeddingNet_65446711656874
// MI455X (gfx1250) — compile-verified
//
#include <hip/hip_runtime.h>

typedef __attribute__((ext_vector_type(2))) float v2f;
typedef __attribute__((ext_vector_type(8))) float v8f;

#define SEL0   46
#define SEL1   92
#define NNEI   138
#define NDESC  552
#define TILES0 3
#define NTILES 9   // 3 tiles (type0, 46->48) + 6 tiles (type1, 92->96)

// gfx1250 hardware tanh (V_TANH_F32, TRANS pipe) — one instruction instead of
// the ~40-op branchy libm expansion; co-executes with the WMMA/XDL pipe.
__device__ __forceinline__ float htanh(float x) {
#if __has_builtin(__builtin_amdgcn_tanh_f32)
    return __builtin_amdgcn_tanh_f32(x);
#elif __has_builtin(__builtin_amdgcn_tanhf)
    return __builtin_amdgcn_tanhf(x);
#else
    return tanhf(x);
#endif
}

__launch_bounds__(128, 2)
__global__ void embed_net_kernel(
    const float* __restrict__ dmatrix,
    const float* __restrict__ w00, const float* __restrict__ b00,
    const float* __restrict__ w01, const float* __restrict__ b01,
    const float* __restrict__ w02, const float* __restrict__ b02,
    const float* __restrict__ w10, const float* __restrict__ b10,
    const float* __restrict__ w11, const float* __restrict__ b11,
    const float* __restrict__ w12, const float* __restrict__ b12,
    float* __restrict__ out, int natoms)
{
    __shared__ float sW0[2][24],    sB0[2][24];
    __shared__ float sW1[2][24*48], sB1[2][48];
    __shared__ float sW2[2][48*96], sB2[2][96];
    __shared__ float dmat[NDESC];
    __shared__ float h0buf[4][16][24];
    __shared__ float h1buf[4][16][48];
    __shared__ float accW[4][4*96];   // per-wave accumulator (deterministic, no atomics)

    const int tid  = threadIdx.x;
    const int atom = blockIdx.x;

    // ---- cooperative LDS staging -------------------------------------------
    for (int i = tid; i < 24; i += 128) {
        sW0[0][i] = w00[i];  sW0[1][i] = w10[i];
        sB0[0][i] = b00[i];  sB0[1][i] = b10[i];
    }
    for (int i = tid; i < 24*48; i += 128) { sW1[0][i] = w01[i]; sW1[1][i] = w11[i]; }
    for (int i = tid; i < 48;    i += 128) { sB1[0][i] = b01[i]; sB1[1][i] = b11[i]; }
    for (int i = tid; i < 48*96; i += 128) { sW2[0][i] = w02[i]; sW2[1][i] = w12[i]; }
    for (int i = tid; i < 96;    i += 128) { sB2[0][i] = b02[i]; sB2[1][i] = b12[i]; }

    const float4* dm4 = (const float4*)(dmatrix + (size_t)atom * NDESC); // 552 f32 = 138 f4
    for (int i = tid; i < NDESC/4; i += 128) ((float4*)dmat)[i] = dm4[i];

    for (int i = tid; i < 4*4*96; i += 128) ((float*)accW)[i] = 0.0f;

    // prefetch next atom's descriptor row while we compute (speculative)
    if (atom + 1 < natoms)
        __builtin_prefetch(dmatrix + (size_t)(atom + 1) * NDESC, 0, 0);

    __syncthreads();

    const int lane  = tid & 31;
    const int wave  = tid >> 5;
    const int lrow  = lane & 15;   // M (A/D) or N (B/D) index within 16
    const int khalf = lane >> 4;   // f32 WMMA 16x16x4: lanes 16-31 carry K=2,3

    // ---- MLP over 16-neighbor row tiles, one wave per tile -----------------
    for (int tile = wave; tile < NTILES; tile += 4) {
        int t, ti, sel, base;
        if (tile < TILES0) { t = 0; ti = tile;          sel = SEL0; base = 0;      }
        else               { t = 1; ti = tile - TILES0; sel = SEL1; base = SEL0*4; }

        // layer0: scalar -> 24, VALU outer product + hw tanh
        for (int idx = lane; idx < 16*24; idx += 32) {
            int r = idx / 24, c = idx % 24;
            int j = ti*16 + r;
            float s = (j < sel) ? dmat[base + j*4] : 0.0f;
            h0buf[wave][r][c] = htanh(s * sW0[t][c] + sB0[t][c]);
        }

        // layer1: [16,24] @ [24,48] via 3 N-tiles x 6 K-steps of wmma f32 16x16x4
        for (int nt = 0; nt < 3; ++nt) {
            v8f c1 = {};
            const int gn = nt*16 + lrow;
            #pragma unroll
            for (int kk = 0; kk < 24; kk += 4) {
                const int k0 = kk + 2*khalf;
                v2f a, b;
                a[0] = h0buf[wave][lrow][k0];
                a[1] = h0buf[wave][lrow][k0 + 1];
                b[0] = sW1[t][(k0    )*48 + gn];
                b[1] = sW1[t][(k0 + 1)*48 + gn];
                c1 = __builtin_amdgcn_wmma_f32_16x16x4_f32(
                        false, a, false, b, (short)0, c1, false, false);
            }
            // h1 = [h0,h0] + tanh(c1 + b1)
            #pragma unroll
            for (int v = 0; v < 8; ++v) {
                const int m = v + 8*khalf;
                h1buf[wave][m][gn] = h0buf[wave][m][gn % 24]
                                   + htanh(c1[v] + sB1[t][gn]);
            }
        }

        // layer2: [16,48] @ [48,96] (6 N-tiles x 12 K-steps) + fused dd^T g
        for (int nt = 0; nt < 6; ++nt) {
            v8f c2 = {};
            const int gn = nt*16 + lrow;
            #pragma unroll
            for (int kk = 0; kk < 48; kk += 4) {
                const int k0 = kk + 2*khalf;
                v2f a, b;
                a[0] = h1buf[wave][lrow][k0];
                a[1] = h1buf[wave][lrow][k0 + 1];
                b[0] = sW2[t][(k0    )*96 + gn];
                b[1] = sW2[t][(k0 + 1)*96 + gn];
                c2 = __builtin_amdgcn_wmma_f32_16x16x4_f32(
                        false, a, false, b, (short)0, c2, false, false);
            }
            // g = [h1,h1] + tanh(c2 + b2); accumulate acc[d][gn] += dd[j][d]*g
            float p0 = 0.f, p1 = 0.f, p2 = 0.f, p3 = 0.f;
            #pragma unroll
            for (int v = 0; v < 8; ++v) {
                const int m = v + 8*khalf;
                const int j = ti*16 + m;
                float g = htanh(c2[v] + sB2[t][gn]) + h1buf[wave][m][gn % 48];
                if (j < sel) {
                    const float* dd = &dmat[base + j*4];
                    p0 += dd[0]*g; p1 += dd[1]*g; p2 += dd[2]*g; p3 += dd[3]*g;
                }
            }
            // combine the two lane-halves (rows 0-7 vs 8-15), wave32 shuffle
            p0 += __shfl_xor(p0, 16);
            p1 += __shfl_xor(p1, 16);
            p2 += __shfl_xor(p2, 16);
            p3 += __shfl_xor(p3, 16);
            if (khalf == 0) {   // one writer per (wave, column)
                accW[wave][0*96 + gn] += p0;
                accW[wave][1*96 + gn] += p1;
                accW[wave][2*96 + gn] += p2;
                accW[wave][3*96 + gn] += p3;
            }
        }
    }
    __syncthreads();

    // ---- reduce per-wave accumulators, normalize ---------------------------
    const float inv = 1.0f / (float)NNEI;
    for (int i = tid; i < 4*96; i += 128)
        accW[0][i] = (accW[0][i] + accW[1][i] + accW[2][i] + accW[3][i]) * inv;
    __syncthreads();

    // ---- res[e][k] = sum_d xs[d][e] * xs[d][k], k < 8 ----------------------
    float* xs = accW[0];
    for (int idx = tid; idx < 96*8; idx += 128) {
        const int e = idx >> 3, k = idx & 7;
        float r = 0.0f;
        for (int d = 0; d < 4; ++d)
            r += xs[d*96 + e] * xs[d*96 + k];
        out[(size_t)atom * (96*8) + idx] = r;
    }
}

extern "C" void kernel_launch(void* const* d_in, const int* in_sizes, int n_in,
                              void* d_out, int out_size, void* d_ws, size_t ws_size,
                              hipStream_t stream) {
    const float* dmatrix = (const float*)d_in[0];
    const float* w00 = (const float*)d_in[1];
    const float* b00 = (const float*)d_in[2];
    const float* w01 = (const float*)d_in[3];
    const float* b01 = (const float*)d_in[4];
    const float* w02 = (const float*)d_in[5];
    const float* b02 = (const float*)d_in[6];
    const float* w10 = (const float*)d_in[7];
    const float* b10 = (const float*)d_in[8];
    const float* w11 = (const float*)d_in[9];
    const float* b11 = (const float*)d_in[10];
    const float* w12 = (const float*)d_in[11];
    const float* b12 = (const float*)d_in[12];
    float* out = (float*)d_out;

    const int natoms = in_sizes[0] / NDESC;   // 6144
    embed_net_kernel<<<natoms, 128, 0, stream>>>(
        dmatrix, w00, b00, w01, b01, w02, b02,
        w10, b10, w11, b11, w12, b12, out, natoms);
}